// GCNLayer_45586782880363
// MI455X (gfx1250) — compile-verified
//
#include <hip/hip_runtime.h>
#include <hip/hip_bf16.h>

#define N_NODES 50000
#define N_EDGES 1600000
#define C_DIM   128

typedef __attribute__((ext_vector_type(2))) float v2f;
typedef __attribute__((ext_vector_type(8))) float v8f;

// ---------------------------------------------------------------------------
// 1) deg[i] = 1.0f  (self-loop contributes exactly one to every node's degree)
// ---------------------------------------------------------------------------
__global__ __launch_bounds__(256) void gcn_init_deg(float* __restrict__ deg) {
    int i = blockIdx.x * blockDim.x + threadIdx.x;
    if (i < N_NODES) deg[i] = 1.0f;
}

// ---------------------------------------------------------------------------
// 2) deg[col[e]] += 1 for every edge (fp32 L2 atomics)
// ---------------------------------------------------------------------------
__global__ __launch_bounds__(256) void gcn_degree(const long long* __restrict__ ei,
                                                  float* __restrict__ deg) {
    int e = blockIdx.x * blockDim.x + threadIdx.x;
    if (e < N_EDGES) {
        int col = (int)ei[(long long)N_EDGES + e];
        atomicAdd(&deg[col], 1.0f);
    }
}

// ---------------------------------------------------------------------------
// 3) dinv[i] = rsqrt(deg[i])   (deg >= 1 always, no zero guard needed)
// ---------------------------------------------------------------------------
__global__ __launch_bounds__(256) void gcn_rsqrt(const float* __restrict__ deg,
                                                 float* __restrict__ dinv) {
    int i = blockIdx.x * blockDim.x + threadIdx.x;
    if (i < N_NODES) dinv[i] = __frsqrt_rn(deg[i]);
}

// ---------------------------------------------------------------------------
// 4) A[i,c] = dinv[i]^2 * x[i,c]   (self-loop term; also zero-bases A)
// ---------------------------------------------------------------------------
__global__ __launch_bounds__(256) void gcn_selfloop(const float* __restrict__ x,
                                                    const float* __restrict__ dinv,
                                                    float* __restrict__ A) {
    long long idx = (long long)blockIdx.x * blockDim.x + threadIdx.x;
    if (idx < (long long)N_NODES * C_DIM) {
        int node = (int)(idx >> 7);   // /128
        float d  = dinv[node];
        A[idx] = d * d * x[idx];
    }
}

// ---------------------------------------------------------------------------
// 5) edge scatter: one wave32 per edge.
//    lane l handles channels {l, l+32, l+64, l+96}:
//      - gather x[row] : 4 fully-coalesced 128B loads per wave
//      - global_atomic_add_f32 into A[col] : 4 fully-coalesced atomic bursts
// ---------------------------------------------------------------------------
__global__ __launch_bounds__(256) void gcn_scatter(const float* __restrict__ x,
                                                   const long long* __restrict__ ei,
                                                   const float* __restrict__ dinv,
                                                   float* __restrict__ A) {
    int wave = (blockIdx.x * blockDim.x + threadIdx.x) >> 5;
    int lane = threadIdx.x & 31;
    if (wave >= N_EDGES) return;
    int row = (int)ei[wave];
    int col = (int)ei[(long long)N_EDGES + wave];
    float norm = dinv[row] * dinv[col];
    const float* xs = x + (long long)row * C_DIM;
    float*       Ad = A + (long long)col * C_DIM;
#pragma unroll
    for (int k = 0; k < 4; ++k) {
        int c = lane + (k << 5);
        atomicAdd(&Ad[c], norm * xs[c]);
    }
}

// ---------------------------------------------------------------------------
// 6) out = relu(A @ W + b) with native fp32 WMMA (V_WMMA_F32_16X16X4_F32).
//    M = 50000 = 3125 * 16 (exact), N = 128 = 8 * 16 (exact) -> no guards,
//    EXEC is all-ones at every WMMA as the ISA requires.
//    Block = 256 threads = 8 waves; block b owns M-tile b, wave w owns N-tile w.
//    K loop: 128/4 = 32 WMMAs, fragments per ISA 7.12.2:
//      A 16x4 : lane<16 -> (M=lane,  K=kk+0/1), lane>=16 -> (M=lane-16, K=kk+2/3)
//      B 4x16 : lane<16 -> (K=kk+0/1, N=lane), lane>=16 -> (K=kk+2/3, N=lane-16)
//      C 16x16: vgpr r  -> M = 8*(lane>>4) + r, N = lane&15
// ---------------------------------------------------------------------------
__global__ __launch_bounds__(256) void gcn_gemm_wmma(const float* __restrict__ A,
                                                     const float* __restrict__ W,
                                                     const float* __restrict__ bias,
                                                     float* __restrict__ out) {
    const int mtile = blockIdx.x;          // 0..3124
    const int wv    = threadIdx.x >> 5;    // 0..7  -> N-tile
    const int lane  = threadIdx.x & 31;
    const int half  = lane >> 4;           // 0 or 1
    const int l16   = lane & 15;

    const int m0 = mtile * 16;
    const int n0 = wv * 16;
    const int bn = n0 + l16;               // this lane's N column

    const float* Ap = A + (long long)(m0 + l16) * C_DIM;

    v8f c = {};
#pragma unroll
    for (int kk = 0; kk < C_DIM; kk += 4) {
        const int k0 = kk + 2 * half;      // K element pair this lane supplies
        v2f a, b;
        a.x = Ap[k0];
        a.y = Ap[k0 + 1];
        b.x = W[(long long)k0 * C_DIM + bn];
        b.y = W[(long long)(k0 + 1) * C_DIM + bn];
        c = __builtin_amdgcn_wmma_f32_16x16x4_f32(
                /*neg_a=*/false, a, /*neg_b=*/false, b,
                /*c_mod=*/(short)0, c, /*reuse_a=*/false, /*reuse_b=*/false);
    }

    const float bv = bias[bn];
#pragma unroll
    for (int r = 0; r < 8; ++r) {
        const int m = m0 + half * 8 + r;
        float v = c[r] + bv;
        out[(long long)m * C_DIM + bn] = v > 0.0f ? v : 0.0f;
    }
}

// ---------------------------------------------------------------------------
// Launch
// ---------------------------------------------------------------------------
extern "C" void kernel_launch(void* const* d_in, const int* in_sizes, int n_in,
                              void* d_out, int out_size, void* d_ws, size_t ws_size,
                              hipStream_t stream) {
    const float*     x  = (const float*)d_in[0];
    const long long* ei = (const long long*)d_in[1];   // int64 edge_index [2, E]
    const float*     W  = (const float*)d_in[2];
    const float*     b  = (const float*)d_in[3];
    float*           out = (float*)d_out;

    // workspace layout: deg [N] | dinv [N] | A [N*C]  (~26 MB total)
    float* deg  = (float*)d_ws;
    float* dinv = deg + N_NODES;
    float* A    = dinv + N_NODES;

    gcn_init_deg<<<(N_NODES + 255) / 256, 256, 0, stream>>>(deg);
    gcn_degree  <<<(N_EDGES + 255) / 256, 256, 0, stream>>>(ei, deg);
    gcn_rsqrt   <<<(N_NODES + 255) / 256, 256, 0, stream>>>(deg, dinv);
    gcn_selfloop<<<((long long)N_NODES * C_DIM + 255) / 256, 256, 0, stream>>>(x, dinv, A);
    gcn_scatter <<<N_EDGES / 8, 256, 0, stream>>>(x, ei, dinv, A);   // 1 wave per edge
    gcn_gemm_wmma<<<N_NODES / 16, 256, 0, stream>>>(A, W, b, out);   // 3125 blocks
}